// TMSD_63565515981058
// MI455X (gfx1250) — compile-verified
//
#include <hip/hip_runtime.h>
#include <math.h>

// Problem dimensions (fixed by the reference)
#define V_DIM  50000
#define T_DIM  200
#define H_DIM  512
#define E_DIM  300
#define B_DIM  64
#define S_DIM  8
#define BS_DIM 512   // B*S

typedef __attribute__((ext_vector_type(2))) float v2f;
typedef __attribute__((ext_vector_type(8))) float v8f;

// ---------------- block-tile config for the WMMA GEMM ----------------
#define BM 32   // block tile M (2 wave rows of 16)
#define BN 64   // block tile N (4 wave cols of 16)
#define KC 32   // K chunk staged in LDS (8 wmma k-steps of 4)
#define KP (KC + 4)  // padded LDS row stride: 36 dwords, 16B-aligned, conflict-free

#define SK_BLOCKS 256  // persistent Sinkhorn grid (co-resident on >=128 WGPs)

__device__ __forceinline__ float softplusf(float x) {
  return x > 30.f ? x : log1pf(expf(x));
}

// ---------------- block reductions (blockDim.x == 256) ----------------
__device__ __forceinline__ float blk_sum(float v) {
  __shared__ float s[256];
  __syncthreads();
  s[threadIdx.x] = v; __syncthreads();
  for (int o = 128; o > 0; o >>= 1) {
    if ((int)threadIdx.x < o) s[threadIdx.x] += s[threadIdx.x + o];
    __syncthreads();
  }
  return s[0];
}
__device__ __forceinline__ float blk_max(float v) {
  __shared__ float s[256];
  __syncthreads();
  s[threadIdx.x] = v; __syncthreads();
  for (int o = 128; o > 0; o >>= 1) {
    if ((int)threadIdx.x < o) s[threadIdx.x] = fmaxf(s[threadIdx.x], s[threadIdx.x + o]);
    __syncthreads();
  }
  return s[0];
}

// ---------------- device-wide sense barrier (persistent kernel) ----------------
__device__ __forceinline__ void gsync(int* cnt, int* sense, int nb) {
  __syncthreads();
  if (threadIdx.x == 0) {
    __threadfence();
    int s = __hip_atomic_load(sense, __ATOMIC_RELAXED, __HIP_MEMORY_SCOPE_AGENT);
    if (__hip_atomic_fetch_add(cnt, 1, __ATOMIC_ACQ_REL, __HIP_MEMORY_SCOPE_AGENT) == nb - 1) {
      __hip_atomic_store(cnt, 0, __ATOMIC_RELAXED, __HIP_MEMORY_SCOPE_AGENT);
      __hip_atomic_fetch_add(sense, 1, __ATOMIC_ACQ_REL, __HIP_MEMORY_SCOPE_AGENT);
    } else {
      while (__hip_atomic_load(sense, __ATOMIC_ACQUIRE, __HIP_MEMORY_SCOPE_AGENT) == s)
        __builtin_amdgcn_s_sleep(1);
    }
  }
  __syncthreads();
}

// ---------------- operand loaders for the generic GEMM ----------------
// C[m,n] = sum_k A[m,k] * B[n,k]   (A: MxK row-major, B: NxK "weight" layout)
struct PlainA {
  const float* p; int ld, rows, cols;
  __device__ __forceinline__ float operator()(int r, int c) const {
    return (r < rows && c < cols) ? p[(size_t)r * ld + c] : 0.f;
  }
};
struct PlainB {
  static constexpr bool kmajor = false;  // contiguous along k
  const float* p; int ld, rows, cols;
  __device__ __forceinline__ float operator()(int n, int k) const {
    return (n < rows && k < cols) ? p[(size_t)n * ld + k] : 0.f;
  }
};
// beta[t,v] = exp(-cost[t,v] - lse[t]) generated on the fly (B[n=v,k=t])
struct BetaB {
  static constexpr bool kmajor = true;   // contiguous along n (v)
  const float* cost; const float* blse; int rows, cols;  // rows=V, cols=T
  __device__ __forceinline__ float operator()(int n, int k) const {
    return (n < rows && k < cols) ? expf(-cost[(size_t)k * V_DIM + n] - blse[k]) : 0.f;
  }
};

// ---------------- generic f32 WMMA GEMM (16x16x4), LDS-staged ----------------
template <bool ATOMIC, class ALd, class BLd>
__global__ void __launch_bounds__(256)
wmma_gemm_nt(ALd A, BLd B, float* __restrict__ C, int ldc,
             int M, int N, int K, int kslice) {
  __shared__ float As[BM][KP];
  __shared__ float Bs[BN][KP];
  const int tid  = threadIdx.x;
  const int lane = tid & 31;
  const int wave = tid >> 5;            // 8 waves
  const int wm = wave >> 2, wn = wave & 3;
  const int m0 = blockIdx.x * BM;
  const int n0 = blockIdx.y * BN;
  const int kbeg = blockIdx.z * kslice;
  int kend = kbeg + kslice; if (kend > K) kend = K;
  const int half = lane >> 4;           // lanes 16-31: K+2 / M+8 halves (ISA layout)
  const int l15  = lane & 15;

  v8f acc = {};
  for (int k0 = kbeg; k0 < kend; k0 += KC) {
    // ---- stage A (BM x KC) ----
    if ((m0 + BM <= A.rows) && (k0 + KC <= A.cols)) {
      // full tile: one float4 per thread, coalesced 16B loads + b128 LDS stores
      int r = tid >> 3, c = (tid & 7) * 4;
      *(float4*)&As[r][c] = *(const float4*)(A.p + (size_t)(m0 + r) * A.ld + (k0 + c));
    } else {
#pragma unroll
      for (int e = 0; e < (BM * KC) / 256; ++e) {
        int i = tid + e * 256;
        As[i >> 5][i & 31] = A(m0 + (i >> 5), k0 + (i & 31));
      }
    }
    // ---- stage B (BN x KC) ----
    if constexpr (BLd::kmajor) {
      if ((n0 + BN <= B.rows) && (k0 + KC <= B.cols)) {
        // beta fast path: thread owns one k-row, 8 consecutive v: 2x float4 + exp
        int k = tid >> 3, n8 = (tid & 7) * 8;
        const float* src = B.cost + (size_t)(k0 + k) * V_DIM + (n0 + n8);
        float ls = B.blse[k0 + k];
        float4 c0 = *(const float4*)(src);
        float4 c1 = *(const float4*)(src + 4);
        Bs[n8 + 0][k] = expf(-c0.x - ls); Bs[n8 + 1][k] = expf(-c0.y - ls);
        Bs[n8 + 2][k] = expf(-c0.z - ls); Bs[n8 + 3][k] = expf(-c0.w - ls);
        Bs[n8 + 4][k] = expf(-c1.x - ls); Bs[n8 + 5][k] = expf(-c1.y - ls);
        Bs[n8 + 6][k] = expf(-c1.z - ls); Bs[n8 + 7][k] = expf(-c1.w - ls);
      } else {
#pragma unroll
        for (int e = 0; e < (BN * KC) / 256; ++e) {
          int i = tid + e * 256;
          Bs[i & 63][i >> 6] = B(n0 + (i & 63), k0 + (i >> 6));
        }
      }
    } else {
      if ((n0 + BN <= B.rows) && (k0 + KC <= B.cols)) {
#pragma unroll
        for (int e = 0; e < 2; ++e) {
          int i = tid + e * 256;
          int n = i >> 3, c = (i & 7) * 4;
          *(float4*)&Bs[n][c] = *(const float4*)(B.p + (size_t)(n0 + n) * B.ld + (k0 + c));
        }
      } else {
#pragma unroll
        for (int e = 0; e < (BN * KC) / 256; ++e) {
          int i = tid + e * 256;
          Bs[i >> 5][i & 31] = B(n0 + (i >> 5), k0 + (i & 31));
        }
      }
    }
    __syncthreads();
    // ---- preload all fragments, then back-to-back WMMAs (overlap LDS latency) ----
    const int mrow = wm * 16 + l15;
    const int nrow = wn * 16 + l15;
    const int koff = 2 * half;
    v2f af[8], bf[8];
#pragma unroll
    for (int s = 0; s < 8; ++s) {
      af[s].x = As[mrow][4 * s + koff];
      af[s].y = As[mrow][4 * s + koff + 1];
      bf[s].x = Bs[nrow][4 * s + koff];
      bf[s].y = Bs[nrow][4 * s + koff + 1];
    }
#pragma unroll
    for (int s = 0; s < 8; ++s)
      acc = __builtin_amdgcn_wmma_f32_16x16x4_f32(false, af[s], false, bf[s],
                                                  (short)0, acc, false, false);
    __syncthreads();
  }
  // C 16x16: VGPR r holds M=r (lanes 0-15) and M=r+8 (lanes 16-31)
#pragma unroll
  for (int r = 0; r < 8; ++r) {
    int gm = m0 + wm * 16 + r + 8 * half;
    int gn = n0 + wn * 16 + l15;
    if (gm < M && gn < N) {
      if constexpr (ATOMIC) atomicAdd(&C[(size_t)gm * ldc + gn], acc[r]);
      else                  C[(size_t)gm * ldc + gn] = acc[r];
    }
  }
}

// ---------------- elementwise / reduction kernels ----------------
__global__ void zero_kernel(float* p, size_t n) {
  for (size_t i = blockIdx.x * 256ull + threadIdx.x; i < n; i += (size_t)gridDim.x * 256)
    p[i] = 0.f;
}
__global__ void ew_softplus_bias(float* C, const float* b, int M, int N) {
  int n = M * N;
  for (int i = blockIdx.x * 256 + threadIdx.x; i < n; i += gridDim.x * 256)
    C[i] = softplusf(C[i] + b[i % N]);
}
__global__ void ew_bias(float* C, const float* b, int M, int N) {
  int n = M * N;
  for (int i = blockIdx.x * 256 + threadIdx.x; i < n; i += gridDim.x * 256)
    C[i] = C[i] + b[i % N];
}
__global__ void rownorm2(const float* x, float* out, int rows, int cols) {
  int lane = threadIdx.x & 31, wave = threadIdx.x >> 5;
  int r = blockIdx.x * 8 + wave;
  if (r >= rows) return;
  float s = 0.f;
  for (int c = lane; c < cols; c += 32) { float t = x[(size_t)r * cols + c]; s += t * t; }
  for (int o = 16; o > 0; o >>= 1) s += __shfl_xor(s, o);
  if (lane == 0) out[r] = s;
}
__global__ void cost_finish(float* cost, float* Kmat, const float* nt, const float* nw) {
  size_t n = (size_t)T_DIM * V_DIM;
  for (size_t i = blockIdx.x * 256ull + threadIdx.x; i < n; i += (size_t)gridDim.x * 256) {
    int t = (int)(i / V_DIM), v = (int)(i % V_DIM);
    float c = nt[t] + nw[v] - 2.f * cost[i];   // cost buffer held the dot product
    cost[i] = c;
    Kmat[i] = expf(-20.f * c);                 // OT_A = 20
  }
}
__global__ void row_lse(const float* cost, float* blse) {  // lse of -cost rows
  int t = blockIdx.x;
  const float* row = cost + (size_t)t * V_DIM;
  float m = -INFINITY;
  for (int v = threadIdx.x; v < V_DIM; v += 256) m = fmaxf(m, -row[v]);
  m = blk_max(m);
  float s = 0.f;
  for (int v = threadIdx.x; v < V_DIM; v += 256) s += expf(-row[v] - m);
  s = blk_sum(s);
  if (threadIdx.x == 0) blse[t] = m + logf(s);
}
__global__ void bn_cols(const float* X, float* Y, int M, int N) {  // BN over batch dim
  int c = blockIdx.x * 256 + threadIdx.x;
  if (c >= N) return;
  float s = 0.f, q = 0.f;
  for (int r = 0; r < M; ++r) { float x = X[(size_t)r * N + c]; s += x; q += x * x; }
  float mean = s / M, var = q / M - mean * mean;
  float rs = rsqrtf(var + 1e-5f);
  for (int r = 0; r < M; ++r) Y[(size_t)r * N + c] = (X[(size_t)r * N + c] - mean) * rs;
}
__global__ void softmax_rows(const float* X, float* Y, int cols) {
  int r = blockIdx.x;
  const float* row = X + (size_t)r * cols;
  float m = -INFINITY;
  for (int c = threadIdx.x; c < cols; c += 256) m = fmaxf(m, row[c]);
  m = blk_max(m);
  float s = 0.f;
  for (int c = threadIdx.x; c < cols; c += 256) s += expf(row[c] - m);
  s = blk_sum(s);
  float inv = 1.f / s;
  for (int c = threadIdx.x; c < cols; c += 256)
    Y[(size_t)r * cols + c] = expf(row[c] - m) * inv;
}
__global__ void theta_ds_kernel(const float* mu_e, const float* theta_doc, float* out) {
  int r = blockIdx.x, b = r >> 3;            // r = b*S + s
  __shared__ float w[T_DIM];
  for (int t = threadIdx.x; t < T_DIM; t += 256)
    w[t] = mu_e[(size_t)r * T_DIM + t] * theta_doc[(size_t)b * T_DIM + t];
  __syncthreads();
  float m = -INFINITY;
  for (int t = threadIdx.x; t < T_DIM; t += 256) m = fmaxf(m, w[t]);
  m = blk_max(m);
  float s = 0.f;
  for (int t = threadIdx.x; t < T_DIM; t += 256) s += expf(w[t] - m);
  s = blk_sum(s);
  float inv = 1.f / s;
  for (int t = threadIdx.x; t < T_DIM; t += 256)
    out[(size_t)r * T_DIM + t] = expf(w[t] - m) * inv;
}
__global__ void kl_doc_kernel(const float* mu, const float* lv, float* acc) {
  int r = blockIdx.x;
  const float PV = 1.f - 1.f / T_DIM;
  float p = 0.f;
  for (int t = threadIdx.x; t < T_DIM; t += 256) {
    float m = mu[(size_t)r * T_DIM + t], l = lv[(size_t)r * T_DIM + t];
    p += 0.5f * (expf(l) / PV + m * m / PV + logf(PV) - l - 1.f);
  }
  p = blk_sum(p);
  if (threadIdx.x == 0) atomicAdd(acc, p);
}
__global__ void kl_e_kernel(const float* mu, const float* lv, float* acc) {
  int r = blockIdx.x;
  const float A2 = 0.25f;
  float p = 0.f;
  for (int t = threadIdx.x; t < T_DIM; t += 256) {
    float m = mu[(size_t)r * T_DIM + t], l = lv[(size_t)r * T_DIM + t];
    float d = m - 1.f;
    p += 0.5f * (expf(l) / A2 + d * d / A2 + logf(A2) - l - 1.f);
  }
  p = blk_sum(p);
  if (threadIdx.x == 0) atomicAdd(acc, p);
}
__global__ void bn_stats(const float* logits, float* cmean, float* crstd) {
  int c = blockIdx.x * 256 + threadIdx.x;
  if (c >= V_DIM) return;
  float s = 0.f, q = 0.f;
  for (int r = 0; r < BS_DIM; ++r) {
    float x = logits[(size_t)r * V_DIM + c]; s += x; q += x * x;
  }
  float mean = s / BS_DIM, var = q / BS_DIM - mean * mean;
  cmean[c] = mean; crstd[c] = rsqrtf(var + 1e-5f);
}
__global__ void nll_kernel(const float* logits, const float* cmean, const float* crstd,
                           const float* bnb, const float* x_sub, float* acc) {
  int r = blockIdx.x;
  const float* row = logits + (size_t)r * V_DIM;
  float m = -INFINITY;
  for (int v = threadIdx.x; v < V_DIM; v += 256) {
    float val = (row[v] - cmean[v]) * crstd[v] + bnb[v];
    m = fmaxf(m, val);
  }
  m = blk_max(m);
  float se = 0.f;
  for (int v = threadIdx.x; v < V_DIM; v += 256) {
    float val = (row[v] - cmean[v]) * crstd[v] + bnb[v];
    se += expf(val - m);
  }
  se = blk_sum(se);
  float inv = 1.f / se;
  float p = 0.f;
  for (int v = threadIdx.x; v < V_DIM; v += 256) {
    float val = (row[v] - cmean[v]) * crstd[v] + bnb[v];
    float prob = expf(val - m) * inv;
    p += x_sub[(size_t)r * V_DIM + v] * logf(prob + 1e-10f);
  }
  p = blk_sum(p);
  if (threadIdx.x == 0) atomicAdd(acc, -p);
}

// ---------------- persistent Sinkhorn (one launch, device-wide barriers) ----------------
__global__ void __launch_bounds__(256)
sinkhorn_persistent(const float* __restrict__ Kmat, float* u, float* vv,
                    float* err, int* done, int* bcnt, int* bsense) {
  const int nb = gridDim.x;
  const int nthreads = nb * 256;
  const int gtid = blockIdx.x * 256 + threadIdx.x;
  __shared__ float su[T_DIM];
  __shared__ int s_done;
  const float aT = 1.f / T_DIM, bV = 1.f / V_DIM;

  if (gtid < T_DIM) u[gtid] = aT;          // u0 = a
  if (gtid == 0) { *done = 0; *err = 0.f; }
  gsync(bcnt, bsense, nb);

  for (int it = 0; it < 1000; ++it) {
    if (threadIdx.x == 0) s_done = *(volatile int*)done;
    __syncthreads();
    const bool compute = (s_done == 0);    // frozen after convergence (ref semantics)
    if (compute) {
      for (int t = threadIdx.x; t < T_DIM; t += 256) su[t] = u[t];
    }
    __syncthreads();
    if (compute) {
      // v_new = b / (K^T u + eps): columns across the whole grid, coalesced
      for (int v = gtid; v < V_DIM; v += nthreads) {
        float s = 0.f;
#pragma unroll 4
        for (int t = 0; t < T_DIM; ++t) s += Kmat[(size_t)t * V_DIM + v] * su[t];
        vv[v] = bV / (s + 1e-16f);
      }
    }
    gsync(bcnt, bsense, nb);
    if (compute) {
      // u_new = a / (K v_new + eps): one row per block
      for (int t = blockIdx.x; t < T_DIM; t += nb) {
        const float* row = Kmat + (size_t)t * V_DIM;
        float s = 0.f;
        for (int v = threadIdx.x; v < V_DIM; v += 256) {
          if (v + 8192 < V_DIM) __builtin_prefetch(&row[v + 8192], 0, 0);
          s += row[v] * vv[v];
        }
        s = blk_sum(s);
        if (threadIdx.x == 0) u[t] = aT / (s + 1e-16f);
      }
    }
    gsync(bcnt, bsense, nb);
    if ((it % 50) == 49) {                  // ref only consults err when (i+1)%50==0
      if (compute) {
        for (int t = threadIdx.x; t < T_DIM; t += 256) su[t] = u[t];
        __syncthreads();
        float p = 0.f;
        for (int v = gtid; v < V_DIM; v += nthreads) {
          float s = 0.f;
          for (int t = 0; t < T_DIM; ++t) s += Kmat[(size_t)t * V_DIM + v] * su[t];
          p += fabsf(vv[v] * s - bV);       // |v .* (K^T u_new) - b|
        }
        p = blk_sum(p);
        if (threadIdx.x == 0) atomicAdd(err, p);
      }
      gsync(bcnt, bsense, nb);
      if (compute && gtid == 0) {
        if (*err < 0.005f) *done = 1;
        *err = 0.f;
      }
      gsync(bcnt, bsense, nb);
    }
  }
}

__global__ void ot_reduce(const float* Kmat, const float* cost, const float* u,
                          const float* vv, float* acc) {
  size_t n = (size_t)T_DIM * V_DIM;
  float p = 0.f;
  for (size_t i = blockIdx.x * 256ull + threadIdx.x; i < n; i += (size_t)gridDim.x * 256) {
    int t = (int)(i / V_DIM), v = (int)(i % V_DIM);
    p += u[t] * Kmat[i] * vv[v] * cost[i];   // pi * cost
  }
  p = blk_sum(p);
  if (threadIdx.x == 0) atomicAdd(acc, p);
}
__global__ void finalize(const float* acc, float* out) {
  // acc: 0=sum nll_r, 1=sum kl_doc_b, 2=sum kl_e rows, 3=sum pi*cost
  out[0] = acc[0] / (float)BS_DIM + acc[1] / (float)B_DIM +
           acc[2] / (float)BS_DIM + 100.f * acc[3];
}

// ---------------- host-side GEMM launcher ----------------
template <bool ATOMIC, class ALd, class BLd>
static inline void launch_gemm(const ALd& a, const BLd& b, float* C, int ldc,
                               int M, int N, int K, int ksplit, hipStream_t s) {
  int kslice = (((K + ksplit - 1) / ksplit) + KC - 1) / KC * KC;
  dim3 grid((M + BM - 1) / BM, (N + BN - 1) / BN, ksplit);
  wmma_gemm_nt<ATOMIC, ALd, BLd><<<grid, 256, 0, s>>>(a, b, C, ldc, M, N, K, kslice);
}

extern "C" void kernel_launch(void* const* d_in, const int* in_sizes, int n_in,
                              void* d_out, int out_size, void* d_ws, size_t ws_size,
                              hipStream_t stream) {
  (void)in_sizes; (void)n_in; (void)out_size; (void)ws_size;
  const float* x_word = (const float*)d_in[0];
  const float* x_sub  = (const float*)d_in[1];   // [512, 50000] flat
  const float* efc1w  = (const float*)d_in[2];
  const float* efc1b  = (const float*)d_in[3];
  const float* efc2w  = (const float*)d_in[4];
  const float* efc2b  = (const float*)d_in[5];
  const float* emuw   = (const float*)d_in[6];
  const float* emub   = (const float*)d_in[7];
  const float* elvw   = (const float*)d_in[8];
  const float* elvb   = (const float*)d_in[9];
  const float* sfc1w  = (const float*)d_in[10];
  const float* sfc1b  = (const float*)d_in[11];
  const float* sfc2w  = (const float*)d_in[12];
  const float* sfc2b  = (const float*)d_in[13];
  const float* decbnb = (const float*)d_in[14];
  const float* topic  = (const float*)d_in[15];
  const float* word   = (const float*)d_in[16];

  // ---- workspace layout (~188 MB f32) ----
  float* w      = (float*)d_ws;
  float* cost   = w;                                 // [T,V]  (dot -> cost in place)
  float* Kmat   = cost + (size_t)T_DIM * V_DIM;      // [T,V]
  float* logits = Kmat + (size_t)T_DIM * V_DIM;      // [BS,V]
  float* p      = logits + (size_t)BS_DIM * V_DIM;
  float* h1 = p;        p += B_DIM * H_DIM;
  float* h2 = p;        p += B_DIM * H_DIM;
  float* mupre = p;     p += B_DIM * T_DIM;
  float* lvpre = p;     p += B_DIM * T_DIM;
  float* mu_bn = p;     p += B_DIM * T_DIM;
  float* lv_bn = p;     p += B_DIM * T_DIM;
  float* th_doc = p;    p += B_DIM * T_DIM;
  float* hs1 = p;       p += BS_DIM * H_DIM;
  float* hs2 = p;       p += BS_DIM * H_DIM;
  float* mu_e = p;      p += BS_DIM * T_DIM;
  float* lv_e = p;      p += BS_DIM * T_DIM;
  float* th_ds = p;     p += BS_DIM * T_DIM;
  float* accs = p;      p += 8;                      // 0:nll 1:kld 2:kle 3:ot 4:err
  int*   ictl = (int*)p; p += 4;                     // 0:done 1:bar_cnt 2:bar_sense
  size_t zeroN = (size_t)(p - h1);                   // everything above needs zeroing
  float* nt = p;        p += T_DIM;
  float* nw = p;        p += V_DIM;
  float* blse = p;      p += T_DIM;
  float* cmean = p;     p += V_DIM;
  float* crstd = p;     p += V_DIM;
  float* u = p;         p += T_DIM;
  float* vv = p;        p += V_DIM;

  zero_kernel<<<1024, 256, 0, stream>>>(h1, zeroN);

  // ---- cost matrix + K + row logsumexp of -cost ----
  rownorm2<<<(T_DIM + 7) / 8, 256, 0, stream>>>(topic, nt, T_DIM, E_DIM);
  rownorm2<<<(V_DIM + 7) / 8, 256, 0, stream>>>(word, nw, V_DIM, E_DIM);
  launch_gemm<false>(PlainA{topic, E_DIM, T_DIM, E_DIM},
                     PlainB{word, E_DIM, V_DIM, E_DIM},
                     cost, V_DIM, T_DIM, V_DIM, E_DIM, 1, stream);
  cost_finish<<<8192, 256, 0, stream>>>(cost, Kmat, nt, nw);
  row_lse<<<T_DIM, 256, 0, stream>>>(cost, blse);

  // ---- doc encoder ----
  launch_gemm<true>(PlainA{x_word, V_DIM, B_DIM, V_DIM},
                    PlainB{efc1w, V_DIM, H_DIM, V_DIM},
                    h1, H_DIM, B_DIM, H_DIM, V_DIM, 16, stream);
  ew_softplus_bias<<<128, 256, 0, stream>>>(h1, efc1b, B_DIM, H_DIM);
  launch_gemm<true>(PlainA{h1, H_DIM, B_DIM, H_DIM},
                    PlainB{efc2w, H_DIM, H_DIM, H_DIM},
                    h2, H_DIM, B_DIM, H_DIM, H_DIM, 2, stream);
  ew_softplus_bias<<<128, 256, 0, stream>>>(h2, efc2b, B_DIM, H_DIM);
  launch_gemm<true>(PlainA{h2, H_DIM, B_DIM, H_DIM},
                    PlainB{emuw, H_DIM, T_DIM, H_DIM},
                    mupre, T_DIM, B_DIM, T_DIM, H_DIM, 4, stream);
  ew_bias<<<64, 256, 0, stream>>>(mupre, emub, B_DIM, T_DIM);
  launch_gemm<true>(PlainA{h2, H_DIM, B_DIM, H_DIM},
                    PlainB{elvw, H_DIM, T_DIM, H_DIM},
                    lvpre, T_DIM, B_DIM, T_DIM, H_DIM, 4, stream);
  ew_bias<<<64, 256, 0, stream>>>(lvpre, elvb, B_DIM, T_DIM);
  bn_cols<<<1, 256, 0, stream>>>(mupre, mu_bn, B_DIM, T_DIM);
  bn_cols<<<1, 256, 0, stream>>>(lvpre, lv_bn, B_DIM, T_DIM);
  softmax_rows<<<B_DIM, 256, 0, stream>>>(mu_bn, th_doc, T_DIM);
  kl_doc_kernel<<<B_DIM, 256, 0, stream>>>(mu_bn, lv_bn, accs + 1);

  // ---- subdoc adapter ----
  launch_gemm<true>(PlainA{x_sub, V_DIM, BS_DIM, V_DIM},
                    PlainB{sfc1w, V_DIM, H_DIM, V_DIM},
                    hs1, H_DIM, BS_DIM, H_DIM, V_DIM, 4, stream);
  ew_softplus_bias<<<1024, 256, 0, stream>>>(hs1, sfc1b, BS_DIM, H_DIM);
  launch_gemm<false>(PlainA{hs1, H_DIM, BS_DIM, H_DIM},
                     PlainB{sfc2w, H_DIM, H_DIM, H_DIM},
                     hs2, H_DIM, BS_DIM, H_DIM, H_DIM, 1, stream);
  ew_softplus_bias<<<1024, 256, 0, stream>>>(hs2, sfc2b, BS_DIM, H_DIM);
  launch_gemm<true>(PlainA{hs2, H_DIM, BS_DIM, H_DIM},
                    PlainB{emuw, H_DIM, T_DIM, H_DIM},
                    mu_e, T_DIM, BS_DIM, T_DIM, H_DIM, 2, stream);
  ew_bias<<<512, 256, 0, stream>>>(mu_e, emub, BS_DIM, T_DIM);
  launch_gemm<true>(PlainA{hs2, H_DIM, BS_DIM, H_DIM},
                    PlainB{elvw, H_DIM, T_DIM, H_DIM},
                    lv_e, T_DIM, BS_DIM, T_DIM, H_DIM, 2, stream);
  ew_bias<<<512, 256, 0, stream>>>(lv_e, elvb, BS_DIM, T_DIM);
  kl_e_kernel<<<BS_DIM, 256, 0, stream>>>(mu_e, lv_e, accs + 2);
  theta_ds_kernel<<<BS_DIM, 256, 0, stream>>>(mu_e, th_doc, th_ds);

  // ---- decoder: logits = theta_ds @ beta (beta generated on the fly) ----
  launch_gemm<false>(PlainA{th_ds, T_DIM, BS_DIM, T_DIM},
                     BetaB{cost, blse, V_DIM, T_DIM},
                     logits, V_DIM, BS_DIM, V_DIM, T_DIM, 1, stream);
  bn_stats<<<(V_DIM + 255) / 256, 256, 0, stream>>>(logits, cmean, crstd);
  nll_kernel<<<BS_DIM, 256, 0, stream>>>(logits, cmean, crstd, decbnb, x_sub, accs + 0);

  // ---- Sinkhorn: single persistent launch, grid-wide barriers ----
  sinkhorn_persistent<<<SK_BLOCKS, 256, 0, stream>>>(Kmat, u, vv, accs + 4,
                                                     ictl + 0, ictl + 1, ictl + 2);
  ot_reduce<<<4096, 256, 0, stream>>>(Kmat, cost, u, vv, accs + 3);

  finalize<<<1, 1, 0, stream>>>(accs, (float*)d_out);
}